// NaiveTransformerLM_19731079758127
// MI455X (gfx1250) — compile-verified
//
#include <hip/hip_runtime.h>
#include <hip/hip_bf16.h>
#include <math.h>

// Model dims (fixed by the reference)
#define VOCAB_SZ 32000
#define DMODEL   1024
#define NLAYER   6
#define NHEADS   16
#define HEADDIM  64
#define DFF_SZ   4096
#define BATCH    2
#define SEQ      2048
#define MTOK     (BATCH * SEQ)   // 4096 token rows
#define EPSLN    1e-5f

typedef __attribute__((ext_vector_type(2))) float v2f;
typedef __attribute__((ext_vector_type(8))) float v8f;

// D = A(16x4 f32) * B(4x16 f32) + C(16x16 f32), exact fp32 accumulate.
__device__ __forceinline__ v8f wmma4(v2f a, v2f b, v8f c) {
  return __builtin_amdgcn_wmma_f32_16x16x4_f32(
      /*neg_a=*/false, a, /*neg_b=*/false, b,
      /*c_mod=*/(short)0, c, /*reuse_a=*/false, /*reuse_b=*/false);
}

// ---------------------------------------------------------------------------
// x[row,:] = emb[ids[row],:] + pos[row % SEQ,:]
__global__ void embed_kernel(const int* __restrict__ ids,
                             const float* __restrict__ emb,
                             const float* __restrict__ pos,
                             float* __restrict__ x) {
  int row = blockIdx.x;
  int t   = row % SEQ;
  int id  = ids[row];
  const float* e = emb + (size_t)id * DMODEL;
  const float* p = pos + (size_t)t * DMODEL;
  float* xr = x + (size_t)row * DMODEL;
  for (int i = threadIdx.x; i < DMODEL; i += blockDim.x)
    xr[i] = e[i] + p[i];
}

// ---------------------------------------------------------------------------
// Row LayerNorm over DMODEL. 256 threads (8 wave32) per row.
__global__ void ln_kernel(const float* __restrict__ x,
                          const float* __restrict__ g,
                          const float* __restrict__ b,
                          float* __restrict__ y) {
  int row = blockIdx.x;
  const float* xr = x + (size_t)row * DMODEL;
  float* yr = y + (size_t)row * DMODEL;

  float s = 0.f, s2 = 0.f;
  for (int i = threadIdx.x; i < DMODEL; i += 256) {
    float v = xr[i];
    s += v; s2 += v * v;
  }
  for (int off = 16; off > 0; off >>= 1) {
    s  += __shfl_down(s, off, 32);
    s2 += __shfl_down(s2, off, 32);
  }
  __shared__ float sh[2][8];
  int wave = threadIdx.x >> 5, lane = threadIdx.x & 31;
  if (lane == 0) { sh[0][wave] = s; sh[1][wave] = s2; }
  __syncthreads();
  if (threadIdx.x == 0) {
    float ts = 0.f, ts2 = 0.f;
    for (int w = 0; w < 8; ++w) { ts += sh[0][w]; ts2 += sh[1][w]; }
    float mu  = ts  * (1.0f / DMODEL);
    float var = ts2 * (1.0f / DMODEL) - mu * mu;
    sh[0][0] = mu;
    sh[1][0] = rsqrtf(var + EPSLN);
  }
  __syncthreads();
  float mu = sh[0][0], rs = sh[1][0];
  for (int i = threadIdx.x; i < DMODEL; i += 256)
    yr[i] = (xr[i] - mu) * rs * g[i] + b[i];
}

// ---------------------------------------------------------------------------
// C[M,N] = epi( A[M,K] * W[N,K]^T )   (einsum 'mk,nk->mn')
// EPI: 0 plain, 1 +resid, 2 +bias+GELU(exact), 3 +bias+resid,
//      4 plain with non-temporal stores (streaming output, e.g. logits)
// One wave -> 16(M) x 128(N) tile (8 accumulators, 8x A reuse).
// 4 waves/block -> 64(M) x 128(N). Requires M%64==0, N%128==0, K%4==0.
// K-loop software-pipelined: next-step operands load into a second register
// buffer while the 8 independent WMMAs of the current step issue.
template <int EPI>
__global__ __launch_bounds__(128, 1)
void gemm_kernel(const float* __restrict__ A,
                 const float* __restrict__ W,
                 const float* __restrict__ bias,
                 const float* __restrict__ resid,
                 float* __restrict__ C,
                 int M, int N, int K) {
  int lane = threadIdx.x & 31;
  int wave = threadIdx.x >> 5;
  int g  = lane >> 4;     // K sub-group: lanes 0-15 -> K{0,1}, 16-31 -> K{2,3}
  int mr = lane & 15;     // row (A) / col (B,C) within tile
  int m0 = (blockIdx.y * 4 + wave) * 16;
  int n0 = blockIdx.x * 128;

  const float* arow = A + (size_t)(m0 + mr) * K + 2 * g;
  const float* wrow = W + (size_t)(n0 + mr) * K + 2 * g;

  v8f acc[8] = {};

  // prologue: load k-step 0
  v2f a_cur = *(const v2f*)(arow);
  v2f b_cur[8];
#pragma unroll
  for (int j = 0; j < 8; ++j)
    b_cur[j] = *(const v2f*)(wrow + (size_t)j * 16 * K);

  for (int k0 = 0; k0 < K - 4; k0 += 4) {
    // prefetch the weight stream one chunk ahead (global_prefetch_b8)
    __builtin_prefetch((const void*)(wrow + k0 + 132), 0, 1);
    // stage next k-step into fresh registers
    v2f a_nxt = *(const v2f*)(arow + k0 + 4);
    v2f b_nxt[8];
#pragma unroll
    for (int j = 0; j < 8; ++j)
      b_nxt[j] = *(const v2f*)(wrow + (size_t)j * 16 * K + k0 + 4);
    // 8 independent WMMA chains on the current buffers
#pragma unroll
    for (int j = 0; j < 8; ++j)
      acc[j] = wmma4(a_cur, b_cur[j], acc[j]);
    a_cur = a_nxt;
#pragma unroll
    for (int j = 0; j < 8; ++j) b_cur[j] = b_nxt[j];
  }
#pragma unroll
  for (int j = 0; j < 8; ++j)
    acc[j] = wmma4(a_cur, b_cur[j], acc[j]);

#pragma unroll
  for (int j = 0; j < 8; ++j) {
    int n = n0 + j * 16 + mr;
#pragma unroll
    for (int r = 0; r < 8; ++r) {
      int m = m0 + r + 8 * g;
      size_t idx = (size_t)m * N + n;
      float v = acc[j][r];
      if (EPI == 2 || EPI == 3) v += bias[n];
      if (EPI == 2) v = 0.5f * v * (1.0f + erff(v * 0.70710678118654752f));
      if (EPI == 1 || EPI == 3) v += resid[idx];
      if (EPI == 4) __builtin_nontemporal_store(v, &C[idx]);
      else          C[idx] = v;
    }
  }
}

// ---------------------------------------------------------------------------
// Flash-style causal attention. One wave per (b, h, 16-query tile).
// qkv layout: [B, T, 3*D], q at +0, k at +D, v at +2D, head h at +h*HEADDIM.
// o: [B, T, D] (heads re-interleaved).
__global__ __launch_bounds__(32, 1)
void attn_kernel(const float* __restrict__ qkv,
                 float* __restrict__ o) {
  __shared__ __align__(16) float pshr[16 * 16];
  int lane = threadIdx.x;
  int g  = lane >> 4;
  int mr = lane & 15;
  int qt = blockIdx.x;
  int bh = blockIdx.y;
  int b  = bh >> 4;            // NHEADS == 16
  int h  = bh & 15;

  const int ld = 3 * DMODEL;
  const float* qb = qkv + (size_t)b * SEQ * ld + (size_t)h * HEADDIM;
  const float* kb = qb + DMODEL;
  const float* vb = qb + 2 * DMODEL;

  float Mrow[8], Lrow[8];
  v8f oacc[4] = {};
#pragma unroll
  for (int r = 0; r < 8; ++r) { Mrow[r] = -3.0e38f; Lrow[r] = 0.f; }

  // Q tile stays resident in registers for the whole kernel.
  const float* qrow = qb + (size_t)(qt * 16 + mr) * ld + 2 * g;
  v2f qreg[16];
#pragma unroll
  for (int i = 0; i < 16; ++i) qreg[i] = *(const v2f*)(qrow + 4 * i);

  for (int kt = 0; kt <= qt; ++kt) {
    // ---- S tile: two 8-step stages; one 8-entry staging buffer (WAR
    //      dependence caps live registers, avoiding spills) ----
    const float* krow = kb + (size_t)(kt * 16 + mr) * ld + 2 * g;
    v2f kreg[8];
#pragma unroll
    for (int i = 0; i < 8; ++i) kreg[i] = *(const v2f*)(krow + 4 * i);
    v8f s = {};
#pragma unroll
    for (int i = 0; i < 8; ++i) s = wmma4(qreg[i], kreg[i], s);
#pragma unroll
    for (int i = 0; i < 8; ++i) kreg[i] = *(const v2f*)(krow + 4 * (i + 8));
#pragma unroll
    for (int i = 0; i < 8; ++i) s = wmma4(qreg[i + 8], kreg[i], s);

    bool diag = (kt == qt);
    float p[8];
#pragma unroll
    for (int r = 0; r < 8; ++r) {
      float v = s[r] * 0.125f;                   // 1/sqrt(HEADDIM)
      int qrel = r + 8 * g;                      // query row in tile
      v = (diag && (mr > qrel)) ? -3.0e38f : v;  // causal mask (key > query)
      float rm = v;
#pragma unroll
      for (int off = 8; off > 0; off >>= 1)
        rm = fmaxf(rm, __shfl_xor(rm, off, 16));
      float Mnew  = fmaxf(Mrow[r], rm);
      float alpha = expf(Mrow[r] - Mnew);
      float pv    = expf(v - Mnew);
      float rs = pv;
#pragma unroll
      for (int off = 8; off > 0; off >>= 1)
        rs += __shfl_xor(rs, off, 16);
      Lrow[r] = Lrow[r] * alpha + rs;
      Mrow[r] = Mnew;
      p[r] = pv;
#pragma unroll
      for (int j = 0; j < 4; ++j) oacc[j][r] *= alpha;  // rescale O rows
    }

    __syncthreads();               // previous iteration's reads done
#pragma unroll
    for (int r = 0; r < 8; ++r)
      pshr[(r + 8 * g) * 16 + mr] = p[r];   // P in row-major [q][k]
    __syncthreads();

    // ---- O += P(16x16) * V(16xHD): per k-step, stage 4 V operands then
    //      issue 4 independent WMMAs ----
    const float* vt = vb + (size_t)(kt * 16) * ld + mr;
#pragma unroll
    for (int k0 = 0; k0 < 16; k0 += 4) {
      v2f a = *(const v2f*)(&pshr[mr * 16 + k0 + 2 * g]);
      v2f bv[4];
#pragma unroll
      for (int j = 0; j < 4; ++j) {
        bv[j].x = vt[(size_t)(k0 + 2 * g) * ld + j * 16];
        bv[j].y = vt[(size_t)(k0 + 2 * g + 1) * ld + j * 16];
      }
#pragma unroll
      for (int j = 0; j < 4; ++j) oacc[j] = wmma4(a, bv[j], oacc[j]);
    }
  }

  // normalize and write out
#pragma unroll
  for (int r = 0; r < 8; ++r) {
    float inv = 1.0f / Lrow[r];
#pragma unroll
    for (int j = 0; j < 4; ++j) oacc[j][r] *= inv;
  }
  float* ob = o + (size_t)b * SEQ * DMODEL + (size_t)h * HEADDIM;
#pragma unroll
  for (int j = 0; j < 4; ++j)
#pragma unroll
    for (int r = 0; r < 8; ++r)
      ob[(size_t)(qt * 16 + r + 8 * g) * DMODEL + j * 16 + mr] = oacc[j][r];
}

// ---------------------------------------------------------------------------
extern "C" void kernel_launch(void* const* d_in, const int* in_sizes, int n_in,
                              void* d_out, int out_size, void* d_ws, size_t ws_size,
                              hipStream_t stream) {
  (void)in_sizes; (void)n_in; (void)out_size; (void)ws_size;
  const int*   ids    = (const int*)  d_in[0];
  const float* emb    = (const float*)d_in[1];
  const float* pos    = (const float*)d_in[2];
  const float* ln1_g  = (const float*)d_in[3];
  const float* ln1_b  = (const float*)d_in[4];
  const float* ln2_g  = (const float*)d_in[5];
  const float* ln2_b  = (const float*)d_in[6];
  const float* qkv_w  = (const float*)d_in[7];
  const float* out_w  = (const float*)d_in[8];
  const float* mlp_w1 = (const float*)d_in[9];
  const float* mlp_b1 = (const float*)d_in[10];
  const float* mlp_w2 = (const float*)d_in[11];
  const float* mlp_b2 = (const float*)d_in[12];
  const float* lnf_g  = (const float*)d_in[13];
  const float* lnf_b  = (const float*)d_in[14];
  float* out = (float*)d_out;

  // Workspace layout (floats): needs MTOK*(3*D + 3D + DFF) = 41.9M floats ~ 168MB
  float* ws   = (float*)d_ws;
  float* x    = ws;                                  // [MTOK, D] residual stream
  float* xn   = x    + (size_t)MTOK * DMODEL;        // [MTOK, D] normed
  float* qkvb = xn   + (size_t)MTOK * DMODEL;        // [MTOK, 3D]
  float* att  = qkvb + (size_t)MTOK * 3 * DMODEL;    // [MTOK, D]
  float* hbuf = att  + (size_t)MTOK * DMODEL;        // [MTOK, DFF]

  embed_kernel<<<MTOK, 256, 0, stream>>>(ids, emb, pos, x);

  for (int l = 0; l < NLAYER; ++l) {
    // attention block
    ln_kernel<<<MTOK, 256, 0, stream>>>(x, ln1_g + l * DMODEL, ln1_b + l * DMODEL, xn);
    gemm_kernel<0><<<dim3(3 * DMODEL / 128, MTOK / 64), 128, 0, stream>>>(
        xn, qkv_w + (size_t)l * 3 * DMODEL * DMODEL, nullptr, nullptr,
        qkvb, MTOK, 3 * DMODEL, DMODEL);
    attn_kernel<<<dim3(SEQ / 16, BATCH * NHEADS), 32, 0, stream>>>(qkvb, att);
    gemm_kernel<1><<<dim3(DMODEL / 128, MTOK / 64), 128, 0, stream>>>(
        att, out_w + (size_t)l * DMODEL * DMODEL, nullptr, x,
        x, MTOK, DMODEL, DMODEL);

    // MLP block
    ln_kernel<<<MTOK, 256, 0, stream>>>(x, ln2_g + l * DMODEL, ln2_b + l * DMODEL, xn);
    gemm_kernel<2><<<dim3(DFF_SZ / 128, MTOK / 64), 128, 0, stream>>>(
        xn, mlp_w1 + (size_t)l * DFF_SZ * DMODEL, mlp_b1 + (size_t)l * DFF_SZ, nullptr,
        hbuf, MTOK, DFF_SZ, DMODEL);
    gemm_kernel<3><<<dim3(DMODEL / 128, MTOK / 64), 128, 0, stream>>>(
        hbuf, mlp_w2 + (size_t)l * DMODEL * DFF_SZ, mlp_b2 + (size_t)l * DMODEL, x,
        x, MTOK, DMODEL, DFF_SZ);
  }

  // final LN + logits = xn @ emb^T (streaming non-temporal stores)
  ln_kernel<<<MTOK, 256, 0, stream>>>(x, lnf_g, lnf_b, xn);
  gemm_kernel<4><<<dim3(VOCAB_SZ / 128, MTOK / 64), 128, 0, stream>>>(
      xn, emb, nullptr, nullptr, out, MTOK, VOCAB_SZ, DMODEL);
}